// QuanvolutionHybridClassifier_65481071408049
// MI455X (gfx1250) — compile-verified
//
#include <hip/hip_runtime.h>
#include <math.h>

typedef __attribute__((ext_vector_type(2))) float v2f;
typedef __attribute__((ext_vector_type(8))) float v8f;

#define IMG_PER_BLOCK 16
#define PATCHES 196          // 14*14 patches per image
#define FEATS 784            // 196*4 features per image
#define FS 785               // LDS row stride (odd -> conflict-free strided reads)
#define NCLASS 10
#define WPAD_ROWS 16         // head_w padded from 10 to 16 rows

// Stage head_w (10x784) into a zero-padded 16x784 buffer in workspace so the
// WMMA B-operand feed is branch-free (all 32 lanes issue real loads).
__global__ __launch_bounds__(256)
void pad_head_w_kernel(const float* __restrict__ head_w, float* __restrict__ wpad)
{
    const int i = blockIdx.x * blockDim.x + threadIdx.x;
    if (i < WPAD_ROWS * FEATS)
        wpad[i] = (i < NCLASS * FEATS) ? head_w[i] : 0.0f;
}

__global__ __launch_bounds__(256)
void quanv_fused_kernel(const float* __restrict__ x,
                        const float* __restrict__ params,
                        const float* __restrict__ wpad,
                        const float* __restrict__ head_b,
                        float* __restrict__ out)
{
    __shared__ float sFeats[IMG_PER_BLOCK * FS];   // 16 x 784 feature tile (~50 KB)
    __shared__ float sGates[8 * 8];                // 2 layers * 4 wires * 4 complex entries
    __shared__ float sTile[16 * 16];               // logits tile

    const int tid = threadIdx.x;

    // ---- Phase 0: build the 8 Rot gates (batch-invariant) into LDS ----
    if (tid < 8) {
        const int layer = tid >> 2;
        const int w     = tid & 3;
        const float phi   = params[layer * 12 + w * 3 + 0];
        const float theta = params[layer * 12 + w * 3 + 1];
        const float omega = params[layer * 12 + w * 3 + 2];
        const float ct = cosf(0.5f * theta);
        const float st = sinf(0.5f * theta);
        const float ap = 0.5f * (phi + omega);
        const float am = 0.5f * (phi - omega);
        const float cap = cosf(ap), sap = sinf(ap);
        const float cam = cosf(am), sam = sinf(am);
        float* g = &sGates[tid * 8];
        g[0] =  cap * ct;  g[1] = -sap * ct;   // g00 = e^{-i(phi+omega)/2} ct
        g[2] = -cam * st;  g[3] = -sam * st;   // g01 = -e^{+i(phi-omega)/2} st
        g[4] =  cam * st;  g[5] = -sam * st;   // g10 = e^{-i(phi-omega)/2} st
        g[6] =  cap * ct;  g[7] =  sap * ct;   // g11 = e^{+i(phi+omega)/2} ct
    }
    __syncthreads();

    // ---- Phase 1: 4-qubit statevector sim per 2x2 patch, fully in registers ----
    for (int p = tid; p < IMG_PER_BLOCK * PATCHES; p += blockDim.x) {
        const int li    = p / PATCHES;            // local image 0..15
        const int patch = p - li * PATCHES;       // 0..195
        const int pi    = patch / 14;
        const int pj    = patch - pi * 14;
        const int b     = blockIdx.x * IMG_PER_BLOCK + li;
        const float* img = x + (size_t)b * FEATS;

        const float ang0 = img[(2 * pi + 0) * 28 + 2 * pj + 0];
        const float ang1 = img[(2 * pi + 0) * 28 + 2 * pj + 1];
        const float ang2 = img[(2 * pi + 1) * 28 + 2 * pj + 0];
        const float ang3 = img[(2 * pi + 1) * 28 + 2 * pj + 1];

        const float c0 = cosf(0.5f * ang0), s0 = sinf(0.5f * ang0);
        const float c1 = cosf(0.5f * ang1), s1 = sinf(0.5f * ang1);
        const float c2 = cosf(0.5f * ang2), s2 = sinf(0.5f * ang2);
        const float c3 = cosf(0.5f * ang3), s3 = sinf(0.5f * ang3);

        // state index = b0*8 + b1*4 + b2*2 + b3 (wire 0 = MSB)
        float stx[16], sty[16];
        #pragma unroll
        for (int i = 0; i < 16; ++i) {
            float r = ((i & 8) ? s0 : c0) * ((i & 4) ? s1 : c1)
                    * ((i & 2) ? s2 : c2) * ((i & 1) ? s3 : c3);
            stx[i] = r; sty[i] = 0.0f;
        }

        #pragma unroll
        for (int layer = 0; layer < 2; ++layer) {
            #pragma unroll
            for (int w = 0; w < 4; ++w) {
                const float* g = &sGates[(layer * 4 + w) * 8];
                const float g00r = g[0], g00i = g[1], g01r = g[2], g01i = g[3];
                const float g10r = g[4], g10i = g[5], g11r = g[6], g11i = g[7];
                const int m = 8 >> w;
                #pragma unroll
                for (int i0 = 0; i0 < 16; ++i0) {
                    if (i0 & m) continue;
                    const int i1 = i0 | m;
                    const float ar = stx[i0], ai = sty[i0];
                    const float br = stx[i1], bi = sty[i1];
                    stx[i0] = g00r * ar - g00i * ai + g01r * br - g01i * bi;
                    sty[i0] = g00r * ai + g00i * ar + g01r * bi + g01i * br;
                    stx[i1] = g10r * ar - g10i * ai + g11r * br - g11i * bi;
                    sty[i1] = g10r * ai + g10i * ar + g11r * bi + g11i * br;
                }
            }
            // CNOT(0,1): cm=8 tm=4 ; CNOT(2,3): cm=2 tm=1 ; CNOT(1,2): cm=4 tm=2
            #pragma unroll
            for (int c = 0; c < 3; ++c) {
                const int cm = (c == 0) ? 8 : (c == 1) ? 2 : 4;
                const int tm = (c == 0) ? 4 : (c == 1) ? 1 : 2;
                #pragma unroll
                for (int i = 0; i < 16; ++i) {
                    if ((i & cm) && !(i & tm)) {
                        const int j = i | tm;
                        float tr = stx[i]; stx[i] = stx[j]; stx[j] = tr;
                        float ti = sty[i]; sty[i] = sty[j]; sty[j] = ti;
                    }
                }
            }
        }

        float z0 = 0.f, z1 = 0.f, z2 = 0.f, z3 = 0.f;
        #pragma unroll
        for (int i = 0; i < 16; ++i) {
            const float pr = stx[i] * stx[i] + sty[i] * sty[i];
            z0 += (i & 8) ? -pr : pr;
            z1 += (i & 4) ? -pr : pr;
            z2 += (i & 2) ? -pr : pr;
            z3 += (i & 1) ? -pr : pr;
        }
        float* frow = &sFeats[li * FS + patch * 4];
        frow[0] = z0; frow[1] = z1; frow[2] = z2; frow[3] = z3;
    }
    __syncthreads();

    // ---- Phase 2: 16x16x784 head GEMM on wave 0 via V_WMMA_F32_16X16X4_F32 ----
    // A (16x4 f32): lanes 0-15 K={0,1}, lanes 16-31 K={2,3}, M = lane&15
    // B ( 4x16 f32): same halves for K, N = lane&15 (zero-padded rows -> no branches)
    if (tid < 32) {
        const int lane = tid;            // whole wave active -> EXEC all ones
        const int half = lane >> 4;
        const int mrow = lane & 15;
        const v2f* wrow2 = (const v2f*)(wpad + (size_t)mrow * FEATS);
        __builtin_prefetch(wrow2, 0, 1);            // global_prefetch_b8

        v8f acc = {};
        #pragma unroll 4
        for (int k = 0; k < FEATS; k += 4) {
            const int ko = k + 2 * half;
            v2f a;
            a.x = sFeats[mrow * FS + ko];            // -> ds_load_2addr_b32
            a.y = sFeats[mrow * FS + ko + 1];
            const v2f bm = wrow2[ko >> 1];           // -> global_load_b64
            acc = __builtin_amdgcn_wmma_f32_16x16x4_f32(
                false, a, false, bm, (short)0, acc, false, false);
        }
        // C/D layout: vgpr v holds row (v + 8*half), column = lane&15
        #pragma unroll
        for (int v = 0; v < 8; ++v)
            sTile[(v + 8 * half) * 16 + mrow] = acc[v];
    }
    __syncthreads();

    // ---- Phase 3: bias + log_softmax, 1 thread per image row ----
    if (tid < IMG_PER_BLOCK) {
        const int row = tid;
        float l[NCLASS];
        float mx = -3.0e38f;
        #pragma unroll
        for (int j = 0; j < NCLASS; ++j) {
            l[j] = sTile[row * 16 + j] + head_b[j];
            mx = fmaxf(mx, l[j]);
        }
        float s = 0.f;
        #pragma unroll
        for (int j = 0; j < NCLASS; ++j) s += expf(l[j] - mx);
        const float lse = logf(s);
        const int b = blockIdx.x * IMG_PER_BLOCK + row;
        #pragma unroll
        for (int j = 0; j < NCLASS; ++j)
            out[b * NCLASS + j] = l[j] - mx - lse;
    }
}

extern "C" void kernel_launch(void* const* d_in, const int* in_sizes, int n_in,
                              void* d_out, int out_size, void* d_ws, size_t ws_size,
                              hipStream_t stream)
{
    const float* x      = (const float*)d_in[0];  // (B,1,28,28)
    const float* params = (const float*)d_in[1];  // (2,4,3)
    const float* head_w = (const float*)d_in[2];  // (10,784)
    const float* head_b = (const float*)d_in[3];  // (10,)
    float* out  = (float*)d_out;                  // (B,10)
    float* wpad = (float*)d_ws;                   // 16x784 zero-padded weights (49 KB)

    const int B = in_sizes[0] / FEATS;            // 1024
    const int grid = B / IMG_PER_BLOCK;           // 64 blocks, 16 images each

    const int padN = WPAD_ROWS * FEATS;
    pad_head_w_kernel<<<(padN + 255) / 256, 256, 0, stream>>>(head_w, wpad);
    quanv_fused_kernel<<<grid, 256, 0, stream>>>(x, params, wpad, head_b, out);
}